// TwoLayerGAT_34720515621374
// MI455X (gfx1250) — compile-verified
//
#include <hip/hip_runtime.h>
#include <hip/hip_bf16.h>

#define NN   100000
#define EE   1600000
#define IN_F 512
#define HID  4
#define H1   4
#define OUT_F 128
#define NEG_SLOPE 0.2f
#define NTILES (NN / 16)   // 6250, exact

typedef float v2f __attribute__((ext_vector_type(2)));
typedef float v8f __attribute__((ext_vector_type(8)));

// ---- monotone float<->uint encoding for atomic max on floats ----
__device__ __forceinline__ unsigned fenc(float f) {
    unsigned u = __float_as_uint(f);
    return (u >> 31) ? ~u : (u | 0x80000000u);
}
__device__ __forceinline__ float fdec(unsigned u) {
    return (u >> 31) ? __uint_as_float(u & 0x7fffffffu) : __uint_as_float(~u);
}
#define ENC_NEG_INF 0x007FFFFFu   // fenc(-inf)

__device__ __forceinline__ float lrelu(float x) {
    return x > 0.0f ? x : NEG_SLOPE * x;
}

// ============ init: m/denom/rst/out ============
__global__ __launch_bounds__(256) void k_init(
    unsigned* __restrict__ m1, float* __restrict__ denom1, float* __restrict__ rst1,
    unsigned* __restrict__ m2, float* __restrict__ denom2, float* __restrict__ out,
    const float* __restrict__ bias1, const float* __restrict__ bias2) {
    long idx = (long)blockIdx.x * 256 + threadIdx.x;
    if (idx >= (long)NN * OUT_F) return;
    out[idx] = bias2[idx & (OUT_F - 1)];
    if (idx < (long)NN * 16) rst1[idx] = bias1[idx & 15];
    if (idx < (long)NN * 4) { m1[idx] = ENC_NEG_INF; denom1[idx] = 0.0f; }
    if (idx < (long)NN)     { m2[idx] = ENC_NEG_INF; denom2[idx] = 0.0f; }
}

// ============ layer-1 GEMM: h1 = nfeats @ fc1_w (WMMA f32 16x16x4) ============
__global__ __launch_bounds__(256) void k_fc1(
    const float* __restrict__ nfeats, const float* __restrict__ fc1_w,
    float* __restrict__ h1) {
    __shared__ float Wl[IN_F * 16];              // 32 KB: full fc1_w
    for (int i = threadIdx.x; i < IN_F * 16; i += 256) Wl[i] = fc1_w[i];
    __syncthreads();

    const int wave = threadIdx.x >> 5;
    const int lane = threadIdx.x & 31;
    const int tile = blockIdx.x * 8 + wave;
    if (tile >= NTILES) return;                  // wave-uniform exit

    const int half = lane >> 4;                  // 0: K=0,1  1: K=2,3
    const int r    = lane & 15;
    const float* arow = nfeats + ((long)tile * 16 + r) * IN_F;

    v8f c = {};
    for (int k = 0; k < IN_F; k += 4) {
        const int kk = k + 2 * half;
        v2f a; v2f b;
        const v2f* ap = (const v2f*)(arow + kk);  // kk even -> 8B aligned
        a = *ap;
        b.x = Wl[kk * 16 + r];
        b.y = Wl[(kk + 1) * 16 + r];
        c = __builtin_amdgcn_wmma_f32_16x16x4_f32(
                false, a, false, b, (short)0, c, false, false);
    }
    // D layout: M = i + 8*half, N = r
    float* o = h1 + (long)tile * 16 * 16;
    #pragma unroll
    for (int i = 0; i < 8; ++i) o[(i + 8 * half) * 16 + r] = c[i];
}

// ============ el/er layer 1 ============
__global__ __launch_bounds__(256) void k_elr1(
    const float* __restrict__ h1, const float* __restrict__ al,
    const float* __restrict__ ar, float* __restrict__ el, float* __restrict__ er) {
    long idx = (long)blockIdx.x * 256 + threadIdx.x;
    if (idx >= (long)NN * H1) return;
    int n = idx >> 2, h = idx & 3;
    const float* hp = h1 + (long)n * 16 + h * 4;
    float sl = 0.f, sr = 0.f;
    #pragma unroll
    for (int d = 0; d < 4; ++d) { sl += hp[d] * al[h * 4 + d]; sr += hp[d] * ar[h * 4 + d]; }
    el[idx] = sl; er[idx] = sr;
}

// ============ edge passes layer 1 ============
__global__ __launch_bounds__(256) void k_max1(
    const int* __restrict__ src, const int* __restrict__ dst,
    const float* __restrict__ el, const float* __restrict__ er,
    unsigned* __restrict__ m) {
    long idx = (long)blockIdx.x * 256 + threadIdx.x;
    if (idx >= (long)EE * H1) return;
    int e = idx >> 2, h = idx & 3;
    int s = src[e], d = dst[e];
    float t = lrelu(el[s * 4 + h] + er[d * 4 + h]);
    atomicMax(&m[d * 4 + h], fenc(t));
}

__global__ __launch_bounds__(256) void k_sum1(
    const int* __restrict__ src, const int* __restrict__ dst,
    const float* __restrict__ el, const float* __restrict__ er,
    const unsigned* __restrict__ m, float* __restrict__ denom) {
    long idx = (long)blockIdx.x * 256 + threadIdx.x;
    if (idx >= (long)EE * H1) return;
    int e = idx >> 2, h = idx & 3;
    int s = src[e], d = dst[e];
    float t = lrelu(el[s * 4 + h] + er[d * 4 + h]);
    atomicAdd(&denom[d * 4 + h], expf(t - fdec(m[d * 4 + h])));
}

__global__ __launch_bounds__(256) void k_msg1(
    const int* __restrict__ src, const int* __restrict__ dst,
    const float* __restrict__ el, const float* __restrict__ er,
    const unsigned* __restrict__ m, const float* __restrict__ denom,
    const float* __restrict__ h1, float* __restrict__ rst) {
    long idx = (long)blockIdx.x * 256 + threadIdx.x;
    if (idx >= (long)EE * H1) return;
    int e = idx >> 2, h = idx & 3;
    int s = src[e], d = dst[e];
    float t = lrelu(el[s * 4 + h] + er[d * 4 + h]);
    float alpha = expf(t - fdec(m[d * 4 + h])) / denom[d * 4 + h];
    const float* hs = h1 + (long)s * 16 + h * 4;
    float* rd = rst + (long)d * 16 + h * 4;
    #pragma unroll
    for (int dd = 0; dd < 4; ++dd) atomicAdd(&rd[dd], alpha * hs[dd]);
}

// ============ mean over d + relu -> x2 (N,4) ============
__global__ __launch_bounds__(256) void k_mean(
    const float* __restrict__ rst, float* __restrict__ x2) {
    long idx = (long)blockIdx.x * 256 + threadIdx.x;
    if (idx >= (long)NN * H1) return;
    int n = idx >> 2, h = idx & 3;
    const float* rp = rst + (long)n * 16 + h * 4;
    float v = 0.25f * (rp[0] + rp[1] + rp[2] + rp[3]);
    x2[idx] = v > 0.f ? v : 0.f;
}

// ============ layer-2 GEMM: h2 = x2 @ fc2_w, K=4 exactly one WMMA ============
__global__ __launch_bounds__(256) void k_fc2(
    const float* __restrict__ x2, const float* __restrict__ fc2_w,
    float* __restrict__ h2) {
    const int wave = threadIdx.x >> 5;
    const int lane = threadIdx.x & 31;
    const int tile = blockIdx.x * 8 + wave;
    if (tile >= NTILES) return;

    const int half = lane >> 4;
    const int r    = lane & 15;
    const long row = (long)tile * 16 + r;
    v2f a; a.x = x2[row * 4 + 2 * half]; a.y = x2[row * 4 + 2 * half + 1];

    float* o = h2 + (long)tile * 16 * OUT_F;
    for (int t = 0; t < OUT_F / 16; ++t) {
        v2f b;
        b.x = fc2_w[(2 * half) * OUT_F + t * 16 + r];
        b.y = fc2_w[(2 * half + 1) * OUT_F + t * 16 + r];
        v8f c = {};
        c = __builtin_amdgcn_wmma_f32_16x16x4_f32(
                false, a, false, b, (short)0, c, false, false);
        #pragma unroll
        for (int i = 0; i < 8; ++i) o[(i + 8 * half) * OUT_F + t * 16 + r] = c[i];
    }
}

// ============ el/er layer 2 ============
__global__ __launch_bounds__(256) void k_elr2(
    const float* __restrict__ h2, const float* __restrict__ al,
    const float* __restrict__ ar, float* __restrict__ el, float* __restrict__ er) {
    long n = (long)blockIdx.x * 256 + threadIdx.x;
    if (n >= NN) return;
    const float* hp = h2 + n * OUT_F;
    float sl = 0.f, sr = 0.f;
    for (int d = 0; d < OUT_F; ++d) { float v = hp[d]; sl += v * al[d]; sr += v * ar[d]; }
    el[n] = sl; er[n] = sr;
}

// ============ edge passes layer 2 ============
__global__ __launch_bounds__(256) void k_max2(
    const int* __restrict__ src, const int* __restrict__ dst,
    const float* __restrict__ el, const float* __restrict__ er,
    unsigned* __restrict__ m) {
    long e = (long)blockIdx.x * 256 + threadIdx.x;
    if (e >= EE) return;
    float t = lrelu(el[src[e]] + er[dst[e]]);
    atomicMax(&m[dst[e]], fenc(t));
}

__global__ __launch_bounds__(256) void k_sum2(
    const int* __restrict__ src, const int* __restrict__ dst,
    const float* __restrict__ el, const float* __restrict__ er,
    const unsigned* __restrict__ m, float* __restrict__ denom) {
    long e = (long)blockIdx.x * 256 + threadIdx.x;
    if (e >= EE) return;
    int d = dst[e];
    float t = lrelu(el[src[e]] + er[d]);
    atomicAdd(&denom[d], expf(t - fdec(m[d])));
}

__global__ __launch_bounds__(256) void k_alpha2(
    const int* __restrict__ src, const int* __restrict__ dst,
    const float* __restrict__ el, const float* __restrict__ er,
    const unsigned* __restrict__ m, const float* __restrict__ denom,
    float* __restrict__ alpha) {
    long e = (long)blockIdx.x * 256 + threadIdx.x;
    if (e >= EE) return;
    int d = dst[e];
    float t = lrelu(el[src[e]] + er[d]);
    alpha[e] = expf(t - fdec(m[d])) / denom[d];
}

// one wave per edge: 32 lanes x float4 = 128 features
__global__ __launch_bounds__(256) void k_msg2(
    const int* __restrict__ src, const int* __restrict__ dst,
    const float* __restrict__ alpha, const float* __restrict__ h2,
    float* __restrict__ out) {
    long idx = (long)blockIdx.x * 256 + threadIdx.x;
    if (idx >= (long)EE * 32) return;
    long e = idx >> 5; int q = (int)(idx & 31);
    float a = alpha[e];
    const float4 v = ((const float4*)(h2 + (long)src[e] * OUT_F))[q];
    float* o = out + (long)dst[e] * OUT_F + q * 4;
    atomicAdd(&o[0], a * v.x);
    atomicAdd(&o[1], a * v.y);
    atomicAdd(&o[2], a * v.z);
    atomicAdd(&o[3], a * v.w);
}

static inline int blocks_for(long n) { return (int)((n + 255) / 256); }

extern "C" void kernel_launch(void* const* d_in, const int* in_sizes, int n_in,
                              void* d_out, int out_size, void* d_ws, size_t ws_size,
                              hipStream_t stream) {
    const float* nfeats  = (const float*)d_in[0];
    const int*   src     = (const int*)d_in[2];
    const int*   dst     = (const int*)d_in[3];
    const float* fc1_w   = (const float*)d_in[4];
    const float* attn_l1 = (const float*)d_in[5];
    const float* attn_r1 = (const float*)d_in[6];
    const float* bias1   = (const float*)d_in[7];
    const float* fc2_w   = (const float*)d_in[8];
    const float* attn_l2 = (const float*)d_in[9];
    const float* attn_r2 = (const float*)d_in[10];
    const float* bias2   = (const float*)d_in[11];
    float* out = (float*)d_out;

    // ---- workspace carve (floats / uints, all 4-byte) ----
    float* ws = (float*)d_ws;
    size_t off = 0;
    float*    h1     = ws + off; off += (size_t)NN * 16;
    float*    el1    = ws + off; off += (size_t)NN * 4;
    float*    er1    = ws + off; off += (size_t)NN * 4;
    unsigned* m1     = (unsigned*)(ws + off); off += (size_t)NN * 4;
    float*    denom1 = ws + off; off += (size_t)NN * 4;
    float*    rst1   = ws + off; off += (size_t)NN * 16;
    float*    x2     = ws + off; off += (size_t)NN * 4;
    float*    h2     = ws + off; off += (size_t)NN * OUT_F;
    float*    el2    = ws + off; off += (size_t)NN;
    float*    er2    = ws + off; off += (size_t)NN;
    unsigned* m2     = (unsigned*)(ws + off); off += (size_t)NN;
    float*    denom2 = ws + off; off += (size_t)NN;
    float*    alpha2 = ws + off; off += (size_t)EE;

    const int gemm_blocks = (NTILES + 7) / 8;

    k_init<<<blocks_for((long)NN * OUT_F), 256, 0, stream>>>(
        m1, denom1, rst1, m2, denom2, out, bias1, bias2);
    k_fc1<<<gemm_blocks, 256, 0, stream>>>(nfeats, fc1_w, h1);
    k_elr1<<<blocks_for((long)NN * 4), 256, 0, stream>>>(h1, attn_l1, attn_r1, el1, er1);
    k_max1<<<blocks_for((long)EE * 4), 256, 0, stream>>>(src, dst, el1, er1, m1);
    k_sum1<<<blocks_for((long)EE * 4), 256, 0, stream>>>(src, dst, el1, er1, m1, denom1);
    k_msg1<<<blocks_for((long)EE * 4), 256, 0, stream>>>(src, dst, el1, er1, m1, denom1, h1, rst1);
    k_mean<<<blocks_for((long)NN * 4), 256, 0, stream>>>(rst1, x2);
    k_fc2<<<gemm_blocks, 256, 0, stream>>>(x2, fc2_w, h2);
    k_elr2<<<blocks_for(NN), 256, 0, stream>>>(h2, attn_l2, attn_r2, el2, er2);
    k_max2<<<blocks_for(EE), 256, 0, stream>>>(src, dst, el2, er2, m2);
    k_sum2<<<blocks_for(EE), 256, 0, stream>>>(src, dst, el2, er2, m2, denom2);
    k_alpha2<<<blocks_for(EE), 256, 0, stream>>>(src, dst, el2, er2, m2, denom2, alpha2);
    k_msg2<<<blocks_for((long)EE * 32), 256, 0, stream>>>(src, dst, alpha2, h2, out);
}